// SplitTCELoss_28260884807701
// MI455X (gfx1250) — compile-verified
//
#include <hip/hip_runtime.h>
#include <hip/hip_bf16.h>
#include <math.h>

// ---------------------------------------------------------------------------
// SplitTCELoss for MI455X (gfx1250, wave32).
//
// Roofline: GEMMs ~0.8 GFLOP (WMMA f32 path, negligible). Dominant cost is
// 128*6145*256 ~= 201M tanh+fma lanes -> transcendental VALU bound. Total
// working set ~10MB << 192MB L2, so HBM (23.3 TB/s) is irrelevant; optimize
// for VALU/TRANS throughput + guaranteed hardware v_tanh_f32.
// ---------------------------------------------------------------------------

#define N_TOK   128
#define D_DIM   256
#define VTOT    6145      // 6144 vocab rows + 1 tail-target token row
#define SPLIT_V 2048
#define NTOKENS 6144
#define TEMPER  65.0f

typedef __attribute__((ext_vector_type(2))) float v2f;
typedef __attribute__((ext_vector_type(8))) float v8f;

// Hardware tanh. Prefer the clang builtin (compiler handles TRANS hazards);
// otherwise emit v_tanh_f32 directly. The asm form appends a v_nop because
// the ISA requires 1 independent op after a TRANS32 before its result is
// consumed, and the compiler cannot see inside the asm blob (ISA 7.4.3).
__device__ __forceinline__ float fast_tanh(float x) {
#if __has_builtin(__builtin_amdgcn_tanhf)
  return __builtin_amdgcn_tanhf(x);
#elif __has_builtin(__builtin_amdgcn_tanh_f32)
  return __builtin_amdgcn_tanh_f32(x);
#else
  float r;
  asm volatile("v_tanh_f32 %0, %1\n\tv_nop" : "=v"(r) : "v"(x));
  return r;
#endif
}

// ---------------------------------------------------------------------------
// C[M,256] = A[M,256] @ B[256,256] (+ bias). One wave per 16x16 output tile,
// v_wmma_f32_16x16x4_f32, K-loop of 64 steps. Tile guard is wave-uniform so
// EXEC==all-ones at every WMMA (ISA requirement).
//
// f32 16x16x4 fragment layout (ISA 7.12.2):
//   A (16x4):  lane<16: (K=0,K=1) of row M=lane; lane>=16: (K=2,K=3)
//   B (4x16):  VGPR0 = rows K=0 (lanes0-15) / K=2 (lanes16-31); VGPR1 = K=1/K=3
//   C (16x16): VGPR j = row m0+j (lanes0-15), row m0+8+j (lanes16-31)
// ---------------------------------------------------------------------------
__global__ __launch_bounds__(256) void wmma_gemm_bias_kernel(
    const float* __restrict__ A, const float* __restrict__ B,
    const float* __restrict__ bias, float* __restrict__ C, int M) {
  const int lane = threadIdx.x & 31;
  const int wave = threadIdx.x >> 5;
  const int rowTiles = (M + 15) >> 4;
  const int totalTiles = rowTiles * (D_DIM / 16);
  const int tile = blockIdx.x * 8 + wave;
  if (tile >= totalTiles) return;              // uniform across the wave

  const int mt = tile >> 4;                    // D_DIM/16 == 16 column tiles
  const int nt = tile & 15;
  const int m0 = mt << 4;
  const int n0 = nt << 4;

  const int r  = lane & 15;
  const int kh = (lane >> 4) << 1;             // 0 or 2

  int arow = m0 + r;
  if (arow >= M) arow = M - 1;                 // clamp; extra rows never stored
  const float* __restrict__ Arow = A + (size_t)arow * D_DIM;
  const float* __restrict__ Bcol = B + (n0 + r);

  v8f acc = {};
  for (int k0 = 0; k0 < D_DIM; k0 += 4) {
    v2f af, bf;
    af.x = Arow[k0 + kh];
    af.y = Arow[k0 + kh + 1];
    bf.x = Bcol[(size_t)(k0 + kh) * D_DIM];
    bf.y = Bcol[(size_t)(k0 + kh + 1) * D_DIM];
    acc = __builtin_amdgcn_wmma_f32_16x16x4_f32(
        /*neg_a=*/false, af, /*neg_b=*/false, bf,
        /*c_mod=*/(short)0, acc, /*reuse_a=*/false, /*reuse_b=*/false);
  }

  const int col = n0 + r;
  const float bv = bias ? bias[col] : 0.0f;
  const int rbase = m0 + ((lane >> 4) << 3);
#pragma unroll
  for (int j = 0; j < 8; ++j) {
    const int row = rbase + j;
    if (row < M) C[(size_t)row * D_DIM + col] = acc[j] + bv;
  }
}

// ---------------------------------------------------------------------------
// K[n,v] = TEMP / (1 + sum_d (h[n,d] - tanh(EW[v,d] + HU[n,d]))^2)
// One wave32 per (n,v). Each lane owns 8 contiguous d (two float4 loads),
// h/HU stay register-resident across the v-loop; shfl_xor wave reduction.
// ---------------------------------------------------------------------------
__global__ __launch_bounds__(256) void dist_kernel(
    const float* __restrict__ h, const float* __restrict__ HU,
    const float* __restrict__ EW, float* __restrict__ Kl) {
  const int lane = threadIdx.x & 31;
  const int wave = threadIdx.x >> 5;
  const int n = blockIdx.y;

  const float* __restrict__ hr = h  + (size_t)n * D_DIM + lane * 8;
  const float* __restrict__ ur = HU + (size_t)n * D_DIM + lane * 8;
  const float4 h0 = *(const float4*)hr;
  const float4 h1 = *(const float4*)(hr + 4);
  const float4 u0 = *(const float4*)ur;
  const float4 u1 = *(const float4*)(ur + 4);

  const int vstride = gridDim.x * 8;
  for (int v = blockIdx.x * 8 + wave; v < VTOT; v += vstride) {
    const float* __restrict__ er = EW + (size_t)v * D_DIM + lane * 8;
    __builtin_prefetch(er + (size_t)vstride * D_DIM, 0, 0);  // global_prefetch_b8
    const float4 e0 = *(const float4*)er;
    const float4 e1 = *(const float4*)(er + 4);

    float t, d, acc = 0.0f;
    t = fast_tanh(e0.x + u0.x); d = h0.x - t; acc = fmaf(d, d, acc);
    t = fast_tanh(e0.y + u0.y); d = h0.y - t; acc = fmaf(d, d, acc);
    t = fast_tanh(e0.z + u0.z); d = h0.z - t; acc = fmaf(d, d, acc);
    t = fast_tanh(e0.w + u0.w); d = h0.w - t; acc = fmaf(d, d, acc);
    t = fast_tanh(e1.x + u1.x); d = h1.x - t; acc = fmaf(d, d, acc);
    t = fast_tanh(e1.y + u1.y); d = h1.y - t; acc = fmaf(d, d, acc);
    t = fast_tanh(e1.z + u1.z); d = h1.z - t; acc = fmaf(d, d, acc);
    t = fast_tanh(e1.w + u1.w); d = h1.w - t; acc = fmaf(d, d, acc);

#pragma unroll
    for (int off = 16; off > 0; off >>= 1)
      acc += __shfl_xor(acc, off, 32);

    if (lane == 0) Kl[(size_t)n * VTOT + v] = TEMPER / (1.0f + acc);
  }
}

// ---------------------------------------------------------------------------
// Per-row log-softmax over head set {0..2047, 6144} and tail set {2048..6143};
// gather target terms -> per-token loss. One block (256 thr) per row.
// ---------------------------------------------------------------------------
__device__ __forceinline__ float block_reduce(float v, float* red, int tid, bool is_max) {
  red[tid] = v;
  __syncthreads();
#pragma unroll
  for (int s = 128; s > 0; s >>= 1) {
    if (tid < s) red[tid] = is_max ? fmaxf(red[tid], red[tid + s]) : (red[tid] + red[tid + s]);
    __syncthreads();
  }
  float r = red[0];
  __syncthreads();
  return r;
}

__global__ __launch_bounds__(256) void softmax_loss_kernel(
    const float* __restrict__ Kl, const int* __restrict__ targets,
    float* __restrict__ lossArr) {
  __shared__ float red[256];
  const int n = blockIdx.x;
  const int tid = threadIdx.x;
  const float* __restrict__ row = Kl + (size_t)n * VTOT;

  // ---- head: cols 0..2047 plus tail-target col 6144 (2049 entries) ----
  float m = -3.402823466e38f;
  for (int i = tid; i < SPLIT_V + 1; i += 256) {
    const int v = (i < SPLIT_V) ? i : NTOKENS;
    m = fmaxf(m, row[v]);
  }
  const float maxh = block_reduce(m, red, tid, true);
  float s = 0.0f;
  for (int i = tid; i < SPLIT_V + 1; i += 256) {
    const int v = (i < SPLIT_V) ? i : NTOKENS;
    s += __expf(row[v] - maxh);
  }
  const float lseh = __logf(block_reduce(s, red, tid, false));

  // ---- tail: cols 2048..6143 (4096 entries) ----
  m = -3.402823466e38f;
  for (int i = tid; i < NTOKENS - SPLIT_V; i += 256)
    m = fmaxf(m, row[SPLIT_V + i]);
  const float maxt = block_reduce(m, red, tid, true);
  s = 0.0f;
  for (int i = tid; i < NTOKENS - SPLIT_V; i += 256)
    s += __expf(row[SPLIT_V + i] - maxt);
  const float lset = __logf(block_reduce(s, red, tid, false));

  if (tid == 0) {
    const int t = targets[n];
    float loss;
    if (t < SPLIT_V) {
      loss = -(row[t] - maxh - lseh);
    } else {
      const float lp_head_last = row[NTOKENS] - maxh - lseh;  // tail-target col
      const float lp_tail      = row[t]       - maxt - lset;  // col SPLIT+(t-SPLIT) == t
      loss = -(lp_head_last + lp_tail);
    }
    lossArr[n] = loss;
  }
}

__global__ __launch_bounds__(128) void final_reduce_kernel(
    const float* __restrict__ lossArr, float* __restrict__ out) {
  __shared__ float red[128];
  const int tid = threadIdx.x;
  red[tid] = lossArr[tid];
  __syncthreads();
#pragma unroll
  for (int s = 64; s > 0; s >>= 1) {
    if (tid < s) red[tid] += red[tid + s];
    __syncthreads();
  }
  if (tid == 0) out[0] = red[0] * (1.0f / (float)N_TOK);
}

// ---------------------------------------------------------------------------
extern "C" void kernel_launch(void* const* d_in, const int* in_sizes, int n_in,
                              void* d_out, int out_size, void* d_ws, size_t ws_size,
                              hipStream_t stream) {
  const float* h   = (const float*)d_in[0];   // [128,256]
  const int*   tg  = (const int*)  d_in[1];   // [128]
  const float* emb = (const float*)d_in[2];   // [6145,256]
  const float* W   = (const float*)d_in[3];   // [256,256]
  const float* U   = (const float*)d_in[4];   // [256,256]
  const float* b   = (const float*)d_in[5];   // [256]

  float* EW      = (float*)d_ws;                 // 6145*256 f32 = 6.29 MB
  float* HU      = EW + (size_t)VTOT * D_DIM;    // 128*256 f32
  float* Kl      = HU + (size_t)N_TOK * D_DIM;   // 128*6145 f32 = 3.15 MB
  float* lossArr = Kl + (size_t)N_TOK * VTOT;    // 128 f32

  // Phase 1: EW = emb @ W  (no bias);  HU = h @ U + b
  {
    const int tilesEW = ((VTOT + 15) >> 4) * (D_DIM / 16);   // 385*16 = 6160
    wmma_gemm_bias_kernel<<<(tilesEW + 7) / 8, 256, 0, stream>>>(emb, W, nullptr, EW, VTOT);
    const int tilesHU = ((N_TOK + 15) >> 4) * (D_DIM / 16);  // 8*16 = 128
    wmma_gemm_bias_kernel<<<(tilesHU + 7) / 8, 256, 0, stream>>>(h, U, b, HU, N_TOK);
  }

  // Phase 2: polynomial-kernel logits K[n,v]
  {
    dim3 grid(96, N_TOK);   // 96*8 = 768 waves per row, looped over 6145 v's
    dist_kernel<<<grid, 256, 0, stream>>>(h, HU, EW, Kl);
  }

  // Phase 3: head/tail log-softmax + target gather -> per-token loss
  softmax_loss_kernel<<<N_TOK, 256, 0, stream>>>(Kl, tg, lossArr);

  // Phase 4: mean over tokens -> scalar
  final_reduce_kernel<<<1, 128, 0, stream>>>(lossArr, (float*)d_out);
}